// GlobalItemConv_26096221290894
// MI455X (gfx1250) — compile-verified
//
#include <hip/hip_runtime.h>
#include <hip/hip_bf16.h>
#include <stdint.h>

// ---------------- problem constants (match reference) ----------------
#define N_NODES 100000
#define N_EDGES 1600000
#define D_FEAT  128
#define EPB     512                 // edges staged per block chunk (1.6M / 512 = 3125 exactly)
#define NCHUNK  (N_EDGES / EPB)

typedef __attribute__((ext_vector_type(2))) float v2f;
typedef __attribute__((ext_vector_type(8))) float v8f;

// ---------------------------------------------------------------------
// Zero-init h1 (workspace) and d_out (h2 accumulator) with float4 stores
// ---------------------------------------------------------------------
__global__ __launch_bounds__(256) void zero2_kernel(float* __restrict__ a,
                                                    float* __restrict__ b,
                                                    int n4) {
  int i = blockIdx.x * blockDim.x + threadIdx.x;
  if (i < n4) {
    float4 z = make_float4(0.f, 0.f, 0.f, 0.f);
    reinterpret_cast<float4*>(a)[i] = z;
    reinterpret_cast<float4*>(b)[i] = z;
  }
}

// ---------------------------------------------------------------------
// SpMM: hout[r] += val * hin[c], edge-parallel, one wave32 per edge.
// Edge stream (row/col/val) is staged into LDS via CDNA5 async
// global->LDS copies (ASYNCcnt), then broadcast-read by all 8 waves.
// Per lane: one b128 gather + 4 no-return global_atomic_add_f32.
// Both feature arrays are L2-resident (51.2 MB each < 192 MB L2).
// ---------------------------------------------------------------------
__global__ __launch_bounds__(256) void spmm_atomic_kernel(
    const float* __restrict__ vals,
    const int*   __restrict__ row,
    const int*   __restrict__ col,
    const float* __restrict__ hin,
    float*       __restrict__ hout) {
  __shared__ int   s_row[EPB];
  __shared__ int   s_col[EPB];
  __shared__ float s_val[EPB];

  const int t    = threadIdx.x;        // 0..255
  const int wave = t >> 5;             // 0..7
  const int lane = t & 31;

  for (int chunk = blockIdx.x; chunk < NCHUNK; chunk += gridDim.x) {
    const long long ebase = (long long)chunk * EPB;

    // --- async stage: each thread copies 2 edges' worth (8 B) per array ---
    {
      const int ei = t * 2;
      unsigned lds_r = (unsigned)(size_t)&s_row[ei];   // generic LDS ptr: low 32 bits = LDS offset
      unsigned lds_c = (unsigned)(size_t)&s_col[ei];
      unsigned lds_v = (unsigned)(size_t)&s_val[ei];
      unsigned long long g_r = (unsigned long long)(const void*)(row  + ebase + ei);
      unsigned long long g_c = (unsigned long long)(const void*)(col  + ebase + ei);
      unsigned long long g_v = (unsigned long long)(const void*)(vals + ebase + ei);
      asm volatile("global_load_async_to_lds_b64 %0, %1, off" :: "v"(lds_r), "v"(g_r) : "memory");
      asm volatile("global_load_async_to_lds_b64 %0, %1, off" :: "v"(lds_c), "v"(g_c) : "memory");
      asm volatile("global_load_async_to_lds_b64 %0, %1, off" :: "v"(lds_v), "v"(g_v) : "memory");
      asm volatile("s_wait_asynccnt 0" ::: "memory");
    }
    __syncthreads();

    // --- process: wave w handles edges w, w+8, ... (64 iterations) ---
    for (int e = wave; e < EPB; e += 8) {
      const int   r = s_row[e];               // uniform per wave -> LDS broadcast
      const int   c = s_col[e];
      const float v = s_val[e];

      const float4 g = *reinterpret_cast<const float4*>(hin + (size_t)c * D_FEAT + lane * 4);
      float* dst = hout + (size_t)r * D_FEAT + lane * 4;
      atomicAdd(dst + 0, v * g.x);            // global_atomic_add_f32, no return
      atomicAdd(dst + 1, v * g.y);
      atomicAdd(dst + 2, v * g.z);
      atomicAdd(dst + 3, v * g.w);
    }
    __syncthreads();   // protect LDS before next chunk restage
  }
}

// ---------------------------------------------------------------------
// Combine: out = (x + h1 + h2) / 3, exact f32, via chained
// V_WMMA_F32_16X16X4_F32. One wave per 16x16 (node x feat) tile.
//   A (16x4 f32): data fragment; lane L<16 holds rows M=L, K={0,1};
//                 lane L>=16 holds K={2,3}.   (ISA 7.12.2 layout)
//   B (4x16 f32): (1/3)*selection;  B[k][n] = (n == 4g+k)/3
//   C/D (16x16):  VGPR v, lane-half h -> M = v + 8h, N = lane%16
// 3 sources x 4 K-groups = 12 accumulating WMMAs per tile.
// Grid covers tiles exactly -> EXEC all-ones (WMMA requirement).
// ---------------------------------------------------------------------
__global__ __launch_bounds__(256) void combine_wmma_kernel(
    const float* __restrict__ x,
    const float* __restrict__ h1,
    float*       __restrict__ out /* holds h2 on entry, result on exit */) {
  const int gwave = (blockIdx.x * blockDim.x + threadIdx.x) >> 5;
  const int lane  = threadIdx.x & 31;
  const int tiles_n = D_FEAT / 16;            // 8
  const int tm = gwave / tiles_n;
  const int tn = gwave % tiles_n;
  const int node0 = tm * 16;
  const int feat0 = tn * 16;
  const int half  = lane >> 4;                // 0 | 1
  const int l16   = lane & 15;
  const float inv3 = 1.0f / 3.0f;

  v8f acc = {};
  const float* srcs[3] = { out, h1, x };

#pragma unroll
  for (int s = 0; s < 3; ++s) {
    const float* S = srcs[s];
#pragma unroll
    for (int g = 0; g < 4; ++g) {
      // A fragment: rows node0+l16, features feat0 + 4g + (2*half .. 2*half+1)
      const float* ap = S + (size_t)(node0 + l16) * D_FEAT + feat0 + 4 * g + 2 * half;
      v2f a;
      a[0] = ap[0];
      a[1] = ap[1];
      // B fragment: k = j + 2*half, n = l16 ; B[k][n] = (n == 4g+k) * (1/3)
      v2f b;
      b[0] = (l16 == 4 * g + 2 * half + 0) ? inv3 : 0.0f;
      b[1] = (l16 == 4 * g + 2 * half + 1) ? inv3 : 0.0f;
      acc = __builtin_amdgcn_wmma_f32_16x16x4_f32(
          /*neg_a=*/false, a, /*neg_b=*/false, b,
          /*c_mod=*/(short)0, acc, /*reuse_a=*/false, /*reuse_b=*/false);
    }
  }

  // Store D per C/D layout: VGPR v, lane-half -> row v + 8*half, col l16
#pragma unroll
  for (int v = 0; v < 8; ++v) {
    out[(size_t)(node0 + v + 8 * half) * D_FEAT + feat0 + l16] = acc[v];
  }
}

// ---------------------------------------------------------------------
extern "C" void kernel_launch(void* const* d_in, const int* in_sizes, int n_in,
                              void* d_out, int out_size, void* d_ws, size_t ws_size,
                              hipStream_t stream) {
  const float* x    = (const float*)d_in[0];   // [N_NODES * D_FEAT]
  const float* vals = (const float*)d_in[1];   // [N_EDGES]
  const int*   row  = (const int*)  d_in[2];   // [N_EDGES]
  const int*   col  = (const int*)  d_in[3];   // [N_EDGES]
  float* out = (float*)d_out;                  // [N_NODES * D_FEAT]
  float* h1  = (float*)d_ws;                   // 51.2 MB scratch

  const int n4 = (N_NODES * D_FEAT) / 4;       // 3.2M float4
  zero2_kernel<<<(n4 + 255) / 256, 256, 0, stream>>>(h1, out, n4);

  // layer 1: h1 = A @ x
  spmm_atomic_kernel<<<NCHUNK, 256, 0, stream>>>(vals, row, col, x, h1);
  // layer 2: out(h2) = A @ h1
  spmm_atomic_kernel<<<NCHUNK, 256, 0, stream>>>(vals, row, col, h1, out);

  // out = (x + h1 + h2) / 3   — WMMA f32 combine, one wave per 16x16 tile
  const int total_waves = (N_NODES / 16) * (D_FEAT / 16);   // 50000
  combine_wmma_kernel<<<total_waves / 8, 256, 0, stream>>>(x, h1, out);
}